// HierarchicalModel_22969485099167
// MI455X (gfx1250) — compile-verified
//
#include <hip/hip_runtime.h>
#include <hip/hip_bf16.h>

typedef __attribute__((ext_vector_type(2))) float v2f;
typedef __attribute__((ext_vector_type(4))) float v4f;
typedef __attribute__((ext_vector_type(8))) float v8f;

#define HDIM 768
#define KC 64            // K-chunk staged in LDS per iteration
#define NCHUNK (HDIM / KC)
#define XSTRIDE 68       // padded LDS row stride (floats): 68 mod 64 = 4 -> conflict-free A reads
#define NPACK 24576      // packed weight floats: 192 k-steps * 2 tiles * 32 lanes * 2
#define CSTRIDE 20       // epilogue LDS row stride

// ---------------------------------------------------------------------------
// Pack weights into per-lane WMMA B-fragment QUADS:
//   quad q = (k0p*2 + t)*32 + L holds, for lane L / N-tile t / k-pair k0p:
//     [0] = W[n][K0a]   [1] = W[n][K0a+1]     (k-step 2*k0p)
//     [2] = W[n][K0b]   [3] = W[n][K0b+1]     (k-step 2*k0p+1)
//   with n = t*16 + L%16 (cols 17..31 zero-padded), K0a = 8*k0p + 2*(L/16),
//   K0b = K0a + 4.  -> one global_load_b128 feeds 2 WMMA k-steps.
// Biases (17 real + pad) appended at float offset NPACK.
// ---------------------------------------------------------------------------
__global__ void pack_weights(const float* __restrict__ Wa, const float* __restrict__ Wb,
                             const float* __restrict__ Wc1, const float* __restrict__ Wc2,
                             const float* __restrict__ Wc3, const float* __restrict__ Wc4,
                             const float* __restrict__ ba, const float* __restrict__ bb,
                             const float* __restrict__ bc1, const float* __restrict__ bc2,
                             const float* __restrict__ bc3, const float* __restrict__ bc4,
                             float* __restrict__ wpack) {
    int tid = blockIdx.x * blockDim.x + threadIdx.x;     // quad index
    if (tid < (HDIM / 8) * 2 * 32) {                     // 96 * 2 * 32 = 6144
        int L   = tid & 31;
        int t   = (tid >> 5) & 1;
        int k0p = tid >> 6;                              // 0..95
        int n   = t * 16 + (L & 15);                     // combined output column
        int K0a = 8 * k0p + 2 * (L >> 4);
        float w[4] = {0.0f, 0.0f, 0.0f, 0.0f};
        const float* src = nullptr; int rn = 0;
        if (n < 2)       { src = Wa;  rn = n;      }
        else if (n < 6)  { src = Wb;  rn = n - 2;  }
        else if (n < 8)  { src = Wc1; rn = n - 6;  }
        else if (n < 11) { src = Wc2; rn = n - 8;  }
        else if (n < 15) { src = Wc3; rn = n - 11; }
        else if (n < 17) { src = Wc4; rn = n - 15; }
        if (src) {
            w[0] = src[rn * HDIM + K0a];
            w[1] = src[rn * HDIM + K0a + 1];
            w[2] = src[rn * HDIM + K0a + 4];
            w[3] = src[rn * HDIM + K0a + 5];
        }
        wpack[4 * tid + 0] = w[0];
        wpack[4 * tid + 1] = w[1];
        wpack[4 * tid + 2] = w[2];
        wpack[4 * tid + 3] = w[3];
    }
    if (tid < 32) {
        float bv = 0.0f;
        if (tid < 2)       bv = ba[tid];
        else if (tid < 6)  bv = bb[tid - 2];
        else if (tid < 8)  bv = bc1[tid - 6];
        else if (tid < 11) bv = bc2[tid - 8];
        else if (tid < 15) bv = bc3[tid - 11];
        else if (tid < 17) bv = bc4[tid - 15];
        wpack[NPACK + tid] = bv;
    }
}

// ---------------------------------------------------------------------------
// Fused multi-head GEMM + routing epilogue.
// 8 waves/block; each wave independently owns a 16-row tile (no block barriers).
// Double-buffered LDS staging: next chunk's global loads overlap current WMMA.
// ---------------------------------------------------------------------------
__global__ __launch_bounds__(256)
void fused_heads(const float* __restrict__ x, const float* __restrict__ wpack,
                 float* __restrict__ out, int Brows) {
    __shared__ float smem[8 * 2 * 16 * XSTRIDE];         // 69.6 KB, 2 buffers/wave

    const int lane = threadIdx.x & 31;
    const int wave = threadIdx.x >> 5;
    float* buf0 = smem + wave * 2 * 16 * XSTRIDE;
    float* buf1 = buf0 + 16 * XSTRIDE;
    const int tileRow = (blockIdx.x * 8 + wave) * 16;
    if (tileRow >= Brows) return;

    v8f acc0 = {};                                       // C tile, cols 0..15
    v8f acc1 = {};                                       // C tile, cols 16..31 (only 16 used)

    const int rowL  = lane & 15;                         // A-frag row within tile
    const int khalf = (lane >> 4) * 2;                   // A-frag K sub-offset (0 or 2)

    // cooperative-load geometry: idx i*32+lane -> row (idx>>4), float4-col (idx&15)
    const int ldR  = lane >> 4;                          // row parity within pair
    const int ldC4 = lane & 15;

    // ---- prologue: fetch chunk 0 into registers ----
    float4 xr[8];
    #pragma unroll
    for (int i = 0; i < 8; ++i) {
        int r = i * 2 + ldR;
        xr[i] = *((const float4*)(x + (size_t)(tileRow + r) * HDIM) + ldC4);
    }

    for (int c = 0; c < NCHUNK; ++c) {
        float* buf = (c & 1) ? buf1 : buf0;

        // ---- store current chunk's registers into LDS ----
        #pragma unroll
        for (int i = 0; i < 8; ++i) {
            int r = i * 2 + ldR;
            *(float4*)(buf + r * XSTRIDE + ldC4 * 4) = xr[i];
        }

        // ---- issue next chunk's global loads (overlaps WMMA below) ----
        if (c + 1 < NCHUNK) {
            #pragma unroll
            for (int i = 0; i < 8; ++i) {
                int r = i * 2 + ldR;
                xr[i] = *((const float4*)(x + (size_t)(tileRow + r) * HDIM
                                            + (c + 1) * KC) + ldC4);
            }
        }

        asm volatile("s_wait_dscnt 0" ::: "memory");     // cross-lane LDS handoff

        // ---- 16 k-steps of f32 WMMA; B frags arrive as b128 quads ----
        #pragma unroll
        for (int k = 0; k < KC; k += 8) {
            int k0p = (c * KC + k) >> 3;                 // global k-pair index
            const v4f* bq = (const v4f*)wpack + (size_t)k0p * 64 + lane;
            v4f b0q = bq[0];                             // N-tile 0, k-steps k and k+4
            v4f b1q = bq[32];                            // N-tile 1, k-steps k and k+4
            v2f aA = *(const v2f*)(buf + rowL * XSTRIDE + k + khalf);
            v2f aB = *(const v2f*)(buf + rowL * XSTRIDE + k + 4 + khalf);

            v2f b00 = __builtin_shufflevector(b0q, b0q, 0, 1);
            v2f b01 = __builtin_shufflevector(b0q, b0q, 2, 3);
            v2f b10 = __builtin_shufflevector(b1q, b1q, 0, 1);
            v2f b11 = __builtin_shufflevector(b1q, b1q, 2, 3);

            acc0 = __builtin_amdgcn_wmma_f32_16x16x4_f32(
                       false, aA, false, b00, (short)0, acc0, false, false);
            acc1 = __builtin_amdgcn_wmma_f32_16x16x4_f32(
                       false, aA, false, b10, (short)0, acc1, false, false);
            acc0 = __builtin_amdgcn_wmma_f32_16x16x4_f32(
                       false, aB, false, b01, (short)0, acc0, false, false);
            acc1 = __builtin_amdgcn_wmma_f32_16x16x4_f32(
                       false, aB, false, b11, (short)0, acc1, false, false);
        }
    }

    // ---- transpose C tile through LDS: [16 rows][CSTRIDE] ----
    {
        int n0    = lane & 15;
        int rbase = (lane >> 4) * 8;                     // C layout: VGPR v -> M = v + 8*(L/16)
        #pragma unroll
        for (int v = 0; v < 8; ++v) {
            buf0[(rbase + v) * CSTRIDE + n0] = acc0[v];
            if (n0 == 0) buf0[(rbase + v) * CSTRIDE + 16] = acc1[v];
        }
    }
    asm volatile("s_wait_dscnt 0" ::: "memory");

    // ---- routing epilogue: lanes 0..15 each handle one row ----
    if (lane < 16) {
        const int r = tileRow + lane;
        const float* bias = wpack + NPACK;
        float cv[17];
        #pragma unroll
        for (int j = 0; j < 17; ++j)
            cv[j] = buf0[lane * CSTRIDE + j] + bias[j];

        // taskA: active iff argmax != 0  <=>  a1 > a0 (first-max ties -> 0)
        float a0 = cv[0], a1 = cv[1];
        bool active = (a1 > a0);
        out[(size_t)r * 2 + 0] = a0;
        out[(size_t)r * 2 + 1] = a1;

        // taskB: pred_b = first-max argmax over 4 logits
        float tb0 = cv[2], tb1 = cv[3], tb2 = cv[4], tb3 = cv[5];
        int pb = 0; float best = tb0;
        if (tb1 > best) { best = tb1; pb = 1; }
        if (tb2 > best) { best = tb2; pb = 2; }
        if (tb3 > best) { best = tb3; pb = 3; }

        float* ob = out + (size_t)Brows * 2 + (size_t)r * 5;
        ob[0] = 0.0f;
        ob[1] = active ? tb0 : 0.0f;
        ob[2] = active ? tb1 : 0.0f;
        ob[3] = active ? tb2 : 0.0f;
        ob[4] = active ? tb3 : 0.0f;

        // taskC: head sizes (2,3,4,2) -> column->head map
        const int headOf[11] = {0,0, 1,1,1, 2,2,2,2, 3,3};
        float* oc = out + (size_t)Brows * 7 + (size_t)r * 11;
        #pragma unroll
        for (int col = 0; col < 11; ++col) {
            bool m = active && (pb == headOf[col]);
            oc[col] = m ? cv[6 + col] : 0.0f;
        }
    }
}

extern "C" void kernel_launch(void* const* d_in, const int* in_sizes, int n_in,
                              void* d_out, int out_size, void* d_ws, size_t ws_size,
                              hipStream_t stream) {
    const float* x   = (const float*)d_in[0];
    const float* Wa  = (const float*)d_in[1];
    const float* ba  = (const float*)d_in[2];
    const float* Wb  = (const float*)d_in[3];
    const float* bb  = (const float*)d_in[4];
    const float* Wc1 = (const float*)d_in[5];
    const float* bc1 = (const float*)d_in[6];
    const float* Wc2 = (const float*)d_in[7];
    const float* bc2 = (const float*)d_in[8];
    const float* Wc3 = (const float*)d_in[9];
    const float* bc3 = (const float*)d_in[10];
    const float* Wc4 = (const float*)d_in[11];
    const float* bc4 = (const float*)d_in[12];

    float* wpack = (float*)d_ws;                 // needs (24576 + 32) * 4 bytes
    float* out   = (float*)d_out;
    int Brows = in_sizes[0] / HDIM;

    {
        int total = (HDIM / 8) * 2 * 32;         // 6144 pack threads (quads)
        pack_weights<<<(total + 255) / 256, 256, 0, stream>>>(
            Wa, Wb, Wc1, Wc2, Wc3, Wc4, ba, bb, bc1, bc2, bc3, bc4, wpack);
    }
    {
        int blocks = (Brows + 127) / 128;        // 128 rows (8 wave-tiles) per block
        fused_heads<<<blocks, 256, 0, stream>>>(x, wpack, out, Brows);
    }
}